// Controller_85332410237553
// MI455X (gfx1250) — compile-verified
//
#include <hip/hip_runtime.h>
#include <math.h>

// ---------------- problem constants ----------------
#define B_   32
#define C_   1024
#define D_   64
#define R_   4
#define NW_  1
#define EPS_ 1e-6f
#define YSTR 796

// y slice offsets (cumsum of SIZES with NW=1, R=4, D=64)
#define OFF_WV   0    // write_vectors (64)
#define OFF_ER   64   // erase_vectors (64)
#define OFF_FG   128  // free_gate (4)
#define OFF_AG   132  // allocation_gate (1)
#define OFF_WG   133  // write_gate (1)
#define OFF_RM   134  // read_mode (12 = R*NM)
#define OFF_WK   146  // write_keys (64)
#define OFF_WS   210  // write_strengths (1)
#define OFF_RK   211  // read_keys (256)
#define OFF_RS   467  // read_strengths (4)
#define OFF_AS   471  // alloc_strengths (1)
#define OFF_RMK  472  // read_mask (256)
#define OFF_WMK  728  // write_mask (64)
#define OFF_MS   792  // mode_strengths (4)

// workspace layout (float offsets)
#define WS_Y    0u          // 32*796 = 25472
#define WS_WW   25472u      // 32*1024 = 32768
#define WS_MEMN 58240u      // 32*1024*64 = 2097152
#define WS_FWD  2155392u    // 32*4*1024 = 131072
#define WS_BWD  2286464u    // 32*4*1024 = 131072
// total = 2417536 floats ~ 9.7 MB

typedef __attribute__((ext_vector_type(2))) float v2f;
typedef __attribute__((ext_vector_type(8))) float v8f;

__device__ __forceinline__ float sigmoidf_(float x) { return 1.0f / (1.0f + __expf(-x)); }
__device__ __forceinline__ float softplusf_(float x) {
  return (x > 20.0f) ? x : log1pf(__expf(x));
}

__device__ __forceinline__ float block_reduce_sum(float v, float* red, int tid) {
  red[tid] = v; __syncthreads();
  for (int s = 128; s > 0; s >>= 1) {
    if (tid < s) red[tid] += red[tid + s];
    __syncthreads();
  }
  float r = red[0]; __syncthreads();
  return r;
}

__device__ __forceinline__ float block_reduce_max(float v, float* red, int tid) {
  red[tid] = v; __syncthreads();
  for (int s = 128; s > 0; s >>= 1) {
    if (tid < s) red[tid] = fmaxf(red[tid], red[tid + s]);
    __syncthreads();
  }
  float r = red[0]; __syncthreads();
  return r;
}

// ---------------------------------------------------------------------------
// Kernel 1: controller GEMM  y = x @ W_if + b_if  via V_WMMA_F32_16X16X4_F32
// M=32 (2 tiles), N=796 (50 tiles of 16, last partially masked), K=512.
// One wave32 per 16x16 output tile.
// A 16x4 layout: lanes 0-15 -> M=lane,   v0=K0, v1=K1
//                lanes 16-31-> M=lane-16,v0=K2, v1=K3
// B 4x16 layout: v0: K = 0 + 2*half, v1: K = 1 + 2*half, N = lane%16
// C/D layout:    vgpr v: M = v + 8*half, N = lane%16
//
// Out-of-range columns of the last N-tile: WMMA column n only feeds output
// column n, so we load B at a CLAMPED (in-bounds) address unconditionally --
// garbage accumulates only in lanes whose store is masked off. This keeps the
// inner loop free of exec-mask predication (pure load + wmma).
// ---------------------------------------------------------------------------
__global__ __launch_bounds__(32) void controller_gemm_wmma(
    const float* __restrict__ x, const float* __restrict__ W,
    const float* __restrict__ bias, float* __restrict__ y) {
  const int nt   = blockIdx.x;          // 0..49
  const int mt   = blockIdx.y;          // 0..1
  const int lane = threadIdx.x;         // 0..31
  const int half = lane >> 4;           // 0/1
  const int l16  = lane & 15;
  const int arow = mt * 16 + l16;       // A-matrix M row for this lane
  const int bcol = nt * 16 + l16;       // B/C-matrix N column for this lane
  const bool colok = (bcol < YSTR);
  const int bsafe = colok ? bcol : (YSTR - 1);   // clamped, always in-bounds

  const float* __restrict__ xa = x + arow * 512 + 2 * half;
  const float* __restrict__ wb = W + (2 * half) * YSTR + bsafe;

  v8f acc = {};
  for (int k0 = 0; k0 < 512; k0 += 4) {
    v2f a, bf;
    a.x  = xa[k0 + 0];
    a.y  = xa[k0 + 1];
    bf.x = wb[(k0 + 0) * YSTR];
    bf.y = wb[(k0 + 1) * YSTR];
    acc = __builtin_amdgcn_wmma_f32_16x16x4_f32(
        /*neg_a=*/false, a, /*neg_b=*/false, bf,
        /*c_mod=*/(short)0, acc, /*reuse_a=*/false, /*reuse_b=*/false);
  }
  if (colok) {
    float bv = bias[bcol];
#pragma unroll
    for (int v = 0; v < 8; v++) {
      int row = mt * 16 + v + 8 * half;
      y[row * YSTR + bcol] = acc[v] + bv;
    }
  }
}

// ---------------------------------------------------------------------------
// Kernel 2 (one block per batch): usage update, write-content cosine,
// allocation softmax, write_weights, memory_new.
// ---------------------------------------------------------------------------
__global__ __launch_bounds__(256) void write_head_update(
    const float* __restrict__ y, const float* __restrict__ memory,
    const float* __restrict__ usage, const float* __restrict__ prw,
    const float* __restrict__ pww, float* __restrict__ ww_out,
    float* __restrict__ memn_out) {
  const int b = blockIdx.x;
  const int tid = threadIdx.x;
  const float* yb = y + b * YSTR;

  __shared__ float wv[D_], er[D_], wmask[D_], wmk[D_];
  __shared__ float fg[R_];
  __shared__ float sc[4];                 // ag, wg, beta_w, spa
  __shared__ float u_l[C_], sim[C_], a_l[C_];
  __shared__ float red[256];

  if (tid < D_) {
    wv[tid] = yb[OFF_WV + tid];
    er[tid] = sigmoidf_(yb[OFF_ER + tid]);
    float m = sigmoidf_(yb[OFF_WMK + tid]);
    wmask[tid] = m;
    wmk[tid]   = m * yb[OFF_WK + tid];
  }
  if (tid < R_) fg[tid] = sigmoidf_(yb[OFF_FG + tid]);
  if (tid == 0) {
    sc[0] = sigmoidf_(yb[OFF_AG]);      // allocation gate
    sc[1] = sigmoidf_(yb[OFF_WG]);      // write gate
    sc[2] = softplusf_(yb[OFF_WS]);     // write strength beta
    sc[3] = softplusf_(yb[OFF_AS]);     // alloc strength (softplus)
  }
  __syncthreads();
  const float ag = sc[0], wg = sc[1], beta = sc[2], spa = sc[3];

  // key norm
  float knp = (tid < D_) ? wmk[tid] * wmk[tid] : 0.0f;
  float kn = sqrtf(block_reduce_sum(knp, red, tid));

  // usage update (NW=1: ww_eff == pww)
  for (int c = tid; c < C_; c += 256) {
    float u  = usage[b * C_ + c];
    float we = pww[b * C_ + c];
    u = u + (1.0f - u) * we;
    float phi = 1.0f;
#pragma unroll
    for (int r = 0; r < R_; r++)
      phi *= (1.0f - fg[r] * prw[(b * R_ + r) * C_ + c]);
    u_l[c] = u * phi;
  }

  // write-content cosine similarity
  for (int c = tid; c < C_; c += 256) {
    const float* mrow = memory + ((size_t)b * C_ + c) * D_;
    float proj = 0.0f, mn2 = 0.0f;
#pragma unroll 8
    for (int d = 0; d < D_; d++) {
      float mm = wmask[d] * mrow[d];
      proj += wmk[d] * mm;
      mn2  += mm * mm;
    }
    sim[c] = proj / (kn * sqrtf(mn2) + EPS_);
  }
  __syncthreads();

  // softmax over C of beta*sim -> write_content (unnormalized in sim)
  float lmax = -1e30f;
  for (int c = tid; c < C_; c += 256) lmax = fmaxf(lmax, beta * sim[c]);
  float gmax = block_reduce_max(lmax, red, tid);
  float lsum = 0.0f;
  for (int c = tid; c < C_; c += 256) {
    float e = __expf(beta * sim[c] - gmax);
    sim[c] = e; lsum += e;
  }
  float wcsum = block_reduce_sum(lsum, red, tid);

  // allocation softmax over C of (1-u)*spa
  float amax = -1e30f;
  for (int c = tid; c < C_; c += 256) {
    float lg = (1.0f - u_l[c]) * spa;
    a_l[c] = lg; amax = fmaxf(amax, lg);
  }
  float gamax = block_reduce_max(amax, red, tid);
  float asum = 0.0f;
  for (int c = tid; c < C_; c += 256) {
    float e = __expf(a_l[c] - gamax);
    a_l[c] = e; asum += e;
  }
  float gasum = block_reduce_sum(asum, red, tid);

  // write weights
  for (int c = tid; c < C_; c += 256) {
    float wc = sim[c] / wcsum;
    float a  = a_l[c] / gasum;
    float w  = wg * (ag * a + (1.0f - ag) * wc);
    ww_out[b * C_ + c] = w;
    a_l[c] = w;  // reuse for memory update
  }
  __syncthreads();

  // memory_new = memory*(1 - ww*erase) + ww*write_vec   (NW=1)
  const size_t base = (size_t)b * C_ * D_;
  for (int i = tid; i < C_ * D_; i += 256) {
    int c = i >> 6, d = i & 63;
    float w = a_l[c];
    memn_out[base + i] = memory[base + i] * (1.0f - w * er[d]) + w * wv[d];
  }
}

// ---------------------------------------------------------------------------
// Kernel 3: fused link-matrix update + fwd/bwd matvecs (single streaming pass
// over the 134 MB link tensor; link_new never materialized).
// Block = (batch b, 32-row chunk). fwd rows owned by block; bwd via atomics.
// ---------------------------------------------------------------------------
__global__ __launch_bounds__(256) void link_fwd_bwd(
    const float* __restrict__ link, const float* __restrict__ prec,
    const float* __restrict__ prw, const float* __restrict__ ww,
    float* __restrict__ fwd_raw, float* __restrict__ bwd_raw) {
  const int b  = blockIdx.y;
  const int ib = blockIdx.x;       // row chunk 0..31
  const int tid = threadIdx.x;

  __shared__ float rw_s[R_][C_];   // 16 KB
  __shared__ float prec_s[C_], ww_s[C_];
  __shared__ float fwdacc[32][R_];

  for (int i = tid; i < C_; i += 256) {
    prec_s[i] = prec[b * C_ + i];
    ww_s[i]   = ww[b * C_ + i];
#pragma unroll
    for (int r = 0; r < R_; r++) rw_s[r][i] = prw[(b * R_ + r) * C_ + i];
  }
  if (tid < 128) fwdacc[tid >> 2][tid & 3] = 0.0f;
  __syncthreads();

  const int i0 = ib * 32;
  float bacc[R_][4];               // bwd partials for owned columns j = tid + 256*jj
#pragma unroll
  for (int r = 0; r < R_; r++)
#pragma unroll
    for (int jj = 0; jj < 4; jj++) bacc[r][jj] = 0.0f;

  for (int il = 0; il < 32; il++) {
    const int i = i0 + il;
    const float wwi  = ww_s[i];
    const float rwi0 = rw_s[0][i], rwi1 = rw_s[1][i];
    const float rwi2 = rw_s[2][i], rwi3 = rw_s[3][i];
    const float* lrow = link + ((size_t)b * C_ + i) * C_;
    __builtin_prefetch(lrow + C_ + tid, 0, 0);   // stream next row

    float f0 = 0.0f, f1 = 0.0f, f2 = 0.0f, f3 = 0.0f;
#pragma unroll
    for (int jj = 0; jj < 4; jj++) {
      const int j = tid + jj * 256;
      float l  = lrow[j];
      float ln = (1.0f - wwi - ww_s[j]) * l + wwi * prec_s[j];
      if (j == i) ln = 0.0f;                     // zero diagonal
      f0 += rw_s[0][j] * ln;
      f1 += rw_s[1][j] * ln;
      f2 += rw_s[2][j] * ln;
      f3 += rw_s[3][j] * ln;
      bacc[0][jj] += rwi0 * ln;
      bacc[1][jj] += rwi1 * ln;
      bacc[2][jj] += rwi2 * ln;
      bacc[3][jj] += rwi3 * ln;
    }
    // wave-level reduce, then one LDS atomic per wave per (row, r)
    float fv[R_] = {f0, f1, f2, f3};
#pragma unroll
    for (int r = 0; r < R_; r++) {
      float v = fv[r];
      for (int off = 16; off > 0; off >>= 1) v += __shfl_down(v, off, 32);
      if ((tid & 31) == 0) atomicAdd(&fwdacc[il][r], v);
    }
  }
  __syncthreads();

  if (tid < 128) {
    int il = tid >> 2, r = tid & 3;
    fwd_raw[(b * R_ + r) * C_ + (i0 + il)] = fwdacc[il][r];
  }
#pragma unroll
  for (int r = 0; r < R_; r++)
#pragma unroll
    for (int jj = 0; jj < 4; jj++) {
      int j = tid + jj * 256;
      atomicAdd(&bwd_raw[(b * R_ + r) * C_ + j], bacc[r][jj]);
    }
}

// ---------------------------------------------------------------------------
// Kernel 4 (one block per (b,r)): sharpen fwd/bwd, read-content cosine,
// read-mode combine, read_vectors = read_weights @ memory_new.
// ---------------------------------------------------------------------------
__global__ __launch_bounds__(256) void read_heads(
    const float* __restrict__ y, const float* __restrict__ fwd_raw,
    const float* __restrict__ bwd_raw, const float* __restrict__ memn,
    float* __restrict__ out) {
  const int r = blockIdx.x;
  const int b = blockIdx.y;
  const int tid = threadIdx.x;
  const float* yb = y + b * YSTR;

  __shared__ float fwds[C_], bwds[C_], rwl[C_];
  __shared__ float mk[D_], msk[D_];
  __shared__ float scal[5];   // sharpen s, beta, bwd_m, cnt_m, fwd_m
  __shared__ float red[256];

  if (tid < D_) {
    float m = sigmoidf_(yb[OFF_RMK + r * D_ + tid]);
    msk[tid] = m;
    mk[tid]  = m * yb[OFF_RK + r * D_ + tid];
  }
  if (tid == 0) {
    scal[0] = softplusf_(yb[OFF_MS + r]);   // mode_strengths (NW=1)
    scal[1] = softplusf_(yb[OFF_RS + r]);   // read strength
    float m0 = yb[OFF_RM + r * 3 + 0];
    float m1 = yb[OFF_RM + r * 3 + 1];
    float m2 = yb[OFF_RM + r * 3 + 2];
    float mx = fmaxf(m0, fmaxf(m1, m2));
    float e0 = __expf(m0 - mx), e1 = __expf(m1 - mx), e2 = __expf(m2 - mx);
    float es = e0 + e1 + e2;
    scal[2] = e0 / es;   // backward mode
    scal[3] = e1 / es;   // content mode
    scal[4] = e2 / es;   // forward mode
  }
  __syncthreads();
  const float s = scal[0], beta = scal[1];
  const float bm = scal[2], cm = scal[3], fm = scal[4];

  // sharpen fwd / bwd:  p = (a + eps)^softplus(ms), normalized
  float fps = 0.0f, bps = 0.0f;
  for (int c = tid; c < C_; c += 256) {
    float pf = powf(fwd_raw[(b * R_ + r) * C_ + c] + EPS_, s);
    float pb = powf(bwd_raw[(b * R_ + r) * C_ + c] + EPS_, s);
    fwds[c] = pf; bwds[c] = pb;
    fps += pf; bps += pb;
  }
  float fsum = block_reduce_sum(fps, red, tid);
  float bsum = block_reduce_sum(bps, red, tid);

  // read-content cosine on memory_new
  float knp = (tid < D_) ? mk[tid] * mk[tid] : 0.0f;
  float kn = sqrtf(block_reduce_sum(knp, red, tid));
  for (int c = tid; c < C_; c += 256) {
    const float* mrow = memn + ((size_t)b * C_ + c) * D_;
    float proj = 0.0f, mn2 = 0.0f;
#pragma unroll 8
    for (int d = 0; d < D_; d++) {
      float mm = msk[d] * mrow[d];
      proj += mk[d] * mm;
      mn2  += mm * mm;
    }
    rwl[c] = beta * proj / (kn * sqrtf(mn2) + EPS_);
  }
  __syncthreads();
  float lmax = -1e30f;
  for (int c = tid; c < C_; c += 256) lmax = fmaxf(lmax, rwl[c]);
  float gmax = block_reduce_max(lmax, red, tid);
  float lsum = 0.0f;
  for (int c = tid; c < C_; c += 256) {
    float e = __expf(rwl[c] - gmax);
    rwl[c] = e; lsum += e;
  }
  float csum = block_reduce_sum(lsum, red, tid);

  // combine read modes
  for (int c = tid; c < C_; c += 256) {
    rwl[c] = cm * (rwl[c] / csum) + fm * (fwds[c] / fsum) + bm * (bwds[c] / bsum);
  }
  __syncthreads();

  // read_vectors[d] = sum_c rw[c] * memory_new[c,d]
  const int d = tid & 63, g = tid >> 6;   // 4 c-groups of 256
  float acc = 0.0f;
  for (int c = g * 256; c < (g + 1) * 256; c++)
    acc += rwl[c] * memn[((size_t)b * C_ + c) * D_ + d];
  red[tid] = acc;
  __syncthreads();
  if (tid < 128) red[tid] += red[tid + 128];
  __syncthreads();
  if (tid < 64) out[(b * R_ + r) * D_ + tid] = red[tid] + red[tid + 64];
}

// ---------------------------------------------------------------------------
extern "C" void kernel_launch(void* const* d_in, const int* in_sizes, int n_in,
                              void* d_out, int out_size, void* d_ws, size_t ws_size,
                              hipStream_t stream) {
  const float* x      = (const float*)d_in[0];
  const float* W_if   = (const float*)d_in[1];
  const float* b_if   = (const float*)d_in[2];
  const float* memory = (const float*)d_in[3];
  const float* usage  = (const float*)d_in[4];
  const float* link   = (const float*)d_in[5];
  const float* prec   = (const float*)d_in[6];
  const float* prw    = (const float*)d_in[7];
  const float* pww    = (const float*)d_in[8];
  float* out = (float*)d_out;
  float* ws  = (float*)d_ws;

  float* y    = ws + WS_Y;
  float* ww   = ws + WS_WW;
  float* memn = ws + WS_MEMN;
  float* fwd  = ws + WS_FWD;
  float* bwd  = ws + WS_BWD;

  hipMemsetAsync(bwd, 0, (size_t)B_ * R_ * C_ * sizeof(float), stream);
  controller_gemm_wmma<<<dim3(50, 2), 32, 0, stream>>>(x, W_if, b_if, y);
  write_head_update<<<B_, 256, 0, stream>>>(y, memory, usage, prw, pww, ww, memn);
  link_fwd_bwd<<<dim3(32, B_), 256, 0, stream>>>(link, prec, prw, ww, fwd, bwd);
  read_heads<<<dim3(R_, B_), 256, 0, stream>>>(y, fwd, bwd, memn, out);
}